// SpiralFC_11089605559039
// MI455X (gfx1250) — compile-verified
//
#include <hip/hip_runtime.h>

typedef float v2f __attribute__((ext_vector_type(2)));
typedef float v4f __attribute__((ext_vector_type(4)));
typedef float v8f __attribute__((ext_vector_type(8)));

#define B_N   64
#define C_IN  64
#define C_OUT 64
#define H_N   112
#define W_N   112
#define HW    (H_N * W_N)            // 12544
#define TPR   7                      // 112 / 16 tiles per row
#define NTILES (B_N * H_N * TPR)     // 50176 16-pixel tiles
#define BLOCK 256                    // 8 waves (wave32)
#define WPB   8
#define GRID  1024

__global__ __launch_bounds__(BLOCK) void spiralfc_wmma_f32(
    const float* __restrict__ x,        // [B, C_IN, H, W]
    const float* __restrict__ weight,   // [C_IN, C_OUT]
    const float* __restrict__ bias,     // [C_OUT]
    const int*   __restrict__ offsets,  // [C_IN, 2] (dh, dw)
    float*       __restrict__ out)      // [B, C_OUT, H, W]
{
    // LDS: weight repacked as k-pairs -> one B-fragment element = one ds_load_b64
    __shared__ float2 wpair[32 * 64];   // wpair[kp*64+n] = (W[2kp][n], W[2kp+1][n]) : 16 KB
    __shared__ int4   offs4[32];        // (dh0,dw0,dh1,dw1) for channels (2kp, 2kp+1)
    __shared__ float  blds[C_OUT];

    const int tid = threadIdx.x;
    for (int i = tid; i < 32 * 64; i += BLOCK) {
        int kp = i >> 6, n = i & 63;
        wpair[i] = make_float2(weight[(2 * kp) * C_OUT + n],
                               weight[(2 * kp + 1) * C_OUT + n]);
    }
    if (tid < 32) {
        offs4[tid] = make_int4(offsets[4 * tid + 0], offsets[4 * tid + 1],
                               offsets[4 * tid + 2], offsets[4 * tid + 3]);
    }
    if (tid < C_OUT) blds[tid] = bias[tid];
    __syncthreads();

    const int lane = tid & 31;
    const int wave = tid >> 5;
    const int half = lane >> 4;     // K pair {0,1} vs {2,3} (half-wave)
    const int l15  = lane & 15;     // M index (A) / N index (B,C,D)

    // Lane-constant hoists (byte-domain channel bases for the two components)
    const int hb0 = half * 2 * HW * 4;        // bytes: channel +2*half
    const int hb1 = hb0 + HW * 4;             // bytes: channel +2*half+1
    float bv[4];
    int   soff[4];
#pragma unroll
    for (int nt = 0; nt < 4; ++nt) {
        bv[nt]   = blds[nt * 16 + l15];
        soff[nt] = (nt * 16 + l15) * HW + half * 8;   // per-lane store offset (elements)
    }

    const int waveGlobal = blockIdx.x * WPB + wave;
    const int waveStride = gridDim.x * WPB;

    // Wave-uniform tile loop => EXEC stays all-ones (WMMA requirement).
    for (int tile = waveGlobal; tile < NTILES; tile += waveStride) {
        // tile is wave-uniform; tell the compiler so bases land in SGPRs
        const int t  = __builtin_amdgcn_readfirstlane(tile);
        const int wt = t % TPR;
        const int bh = t / TPR;
        const int h  = bh % H_N;
        const int b  = bh / H_N;
        const int w0 = wt * 16;

        const char* xb = (const char*)(x + (size_t)((unsigned)b * (C_IN * HW)));  // SGPR base
        const int w0l = w0 + l15;

        // ---- Gather A fragments: 16 K-steps, 2 channels per lane per step ----
        v2f a[16];
#pragma unroll
        for (int ks = 0; ks < 16; ++ks) {
            const int4 o = offs4[2 * ks + half];   // one ds_load_b128, LDS broadcast
            const int hh0 = h + o.x,   ww0 = w0l + o.y;
            const int hh1 = h + o.z,   ww1 = w0l + o.w;
            const bool v0 = ((unsigned)hh0 < H_N) & ((unsigned)ww0 < W_N);
            const bool v1 = ((unsigned)hh1 < H_N) & ((unsigned)ww1 < W_N);
            // i32 BYTE offsets; clamp with unsigned-min so the offset has a
            // provable constant range -> saddr form (sgpr base + zext(u32)).
            // Invalid lanes read a wrong-but-safe address; value zeroed below.
            const unsigned LIM = (unsigned)(C_IN * HW * 4 - 16 * ks * HW - 4);
            unsigned u0 = (unsigned)(hh0 * (W_N * 4) + ((ww0 << 2) + hb0));
            unsigned u1 = (unsigned)(hh1 * (W_N * 4) + ((ww1 << 2) + hb1));
            u0 = u0 < LIM ? u0 : LIM;              // v_min_u32
            u1 = u1 < LIM ? u1 : LIM;
            // ks-constant channel offset (bytes) folds into the 24-bit ioffset
            float f0 = __builtin_nontemporal_load(
                (const float*)(xb + (size_t)(16u * ks * HW) + (size_t)u0));
            float f1 = __builtin_nontemporal_load(
                (const float*)(xb + (size_t)(16u * ks * HW) + (size_t)u1));
            a[ks][0] = v0 ? f0 : 0.0f;
            a[ks][1] = v1 ? f1 : 0.0f;
        }

        // Uniform output row base (SGPR); per-lane part is soff[nt] (+half*8 inside)
        float* ob = out + (size_t)(((unsigned)b * C_OUT * H_N + (unsigned)h) * W_N + (unsigned)w0);

        // ---- 4 N-tiles of 16 outputs; K=64 as 16 x (K=4) WMMAs each ----
#pragma unroll
        for (int nt = 0; nt < 4; ++nt) {
            v8f acc = {0.f, 0.f, 0.f, 0.f, 0.f, 0.f, 0.f, 0.f};
#pragma unroll
            for (int ks = 0; ks < 16; ++ks) {
                // B 4x16 f32: lanes 0-15 K={0,1}, lanes 16-31 K={2,3}; N = l15
                const float2 bw = wpair[(2 * ks + half) * 64 + nt * 16 + l15];
                v2f bf;
                bf[0] = bw.x;
                bf[1] = bw.y;
                acc = __builtin_amdgcn_wmma_f32_16x16x4_f32(
                    /*neg_a=*/false, a[ks],
                    /*neg_b=*/false, bf,
                    /*c_mod=*/(short)0, acc,
                    /*reuse_a=*/false, /*reuse_b=*/false);
            }

            // C/D layout: VGPR v -> M = v + 8*half; lane's N = l15.
            // Each lane owns 8 consecutive w positions -> two b128 NT stores.
            float* p = ob + soff[nt];
            v4f s0 = {acc[0] + bv[nt], acc[1] + bv[nt], acc[2] + bv[nt], acc[3] + bv[nt]};
            v4f s1 = {acc[4] + bv[nt], acc[5] + bv[nt], acc[6] + bv[nt], acc[7] + bv[nt]};
            __builtin_nontemporal_store(s0, (v4f*)p);
            __builtin_nontemporal_store(s1, (v4f*)(p + 4));
        }
    }
}

extern "C" void kernel_launch(void* const* d_in, const int* in_sizes, int n_in,
                              void* d_out, int out_size, void* d_ws, size_t ws_size,
                              hipStream_t stream) {
    (void)in_sizes; (void)n_in; (void)out_size; (void)d_ws; (void)ws_size;
    const float* x       = (const float*)d_in[0];
    const float* weight  = (const float*)d_in[1];
    const float* bias    = (const float*)d_in[2];
    const int*   offsets = (const int*)d_in[3];
    float*       out     = (float*)d_out;

    spiralfc_wmma_f32<<<GRID, BLOCK, 0, stream>>>(x, weight, bias, offsets, out);
}